// AuthorAccumCitationPopModule_26319559590620
// MI455X (gfx1250) — compile-verified
//
#include <hip/hip_runtime.h>
#include <hip/hip_bf16.h>
#include <math.h>
#include <stdint.h>

#define B_PAPERS 4096
#define E_EDGES  131072
#define EMBED    512
#define HIDDEN   1024
#define MAXSEG   1024

typedef __attribute__((ext_vector_type(2))) float v2f;
typedef __attribute__((ext_vector_type(8))) float v8f;

__device__ __forceinline__ float lrelu_f(float x) { return x > 0.0f ? x : 0.01f * x; }

__device__ __forceinline__ int lowerBound(const int* __restrict__ a, int n, int v) {
    int lo = 0, hi = n;
    while (lo < hi) { int m = (lo + hi) >> 1; if (a[m] < v) lo = m + 1; else hi = m; }
    return lo;
}

// CDNA5 async global->LDS copy (tracked by ASYNCcnt, lands directly in LDS,
// no VGPR staging).  lds_byte_off is the offset within the workgroup's LDS
// allocation (low 32 bits of a generic pointer to a __shared__ object).
__device__ __forceinline__ void async_copy_b128(uint32_t lds_byte_off, const void* gaddr) {
    asm volatile("global_load_async_to_lds_b128 %0, %1, off"
                 :: "v"(lds_byte_off), "v"((uint64_t)(uintptr_t)gaddr)
                 : "memory");
}
__device__ __forceinline__ uint32_t lds_off32(const void* shared_ptr) {
    return (uint32_t)(uintptr_t)shared_ptr;   // addrspace(3) offset in low 32 bits
}

// ---------------------------------------------------------------------------
// Kernel 1: per-paper edge softmax (by accum_citations) + weighted aggregation
// of author_h rows -> mix_in[:, 0:512].  One workgroup per paper; edge_dst is
// sorted so each paper's edges form a contiguous segment found by binary search.
// ---------------------------------------------------------------------------
__global__ __launch_bounds__(256) void edge_softmax_agg(
    const float* __restrict__ author_h, const float* __restrict__ accum_cit,
    const int* __restrict__ edge_src, const int* __restrict__ edge_dst,
    float* __restrict__ mix_in) {
    __shared__ float s_coef[MAXSEG];
    __shared__ int   s_src[MAXSEG];
    __shared__ float s_red[8];

    const int p = blockIdx.x;
    const int t = threadIdx.x;
    const int lane = t & 31, wv = t >> 5;

    const int lo = lowerBound(edge_dst, E_EDGES, p);
    const int hi = lowerBound(edge_dst, E_EDGES, p + 1);
    int len = hi - lo;
    if (len > MAXSEG) len = MAXSEG;

    float* orow = mix_in + (size_t)p * (2 * EMBED);
    if (len <= 0) {  // empty segment: segment_sum == 0
        for (int f = t; f < EMBED; f += 256) orow[f] = 0.0f;
        return;
    }

    // Stage citations + src indices into LDS; track local max.
    float lmax = -3.0e38f;
    for (int e = t; e < len; e += 256) {
        int s = edge_src[lo + e];
        float c = accum_cit[s];
        s_src[e] = s;
        s_coef[e] = c;
        lmax = fmaxf(lmax, c);
    }
    for (int off = 16; off; off >>= 1) lmax = fmaxf(lmax, __shfl_down(lmax, off, 32));
    if (lane == 0) s_red[wv] = lmax;
    __syncthreads();
    if (t == 0) { float m = s_red[0]; for (int i = 1; i < 8; ++i) m = fmaxf(m, s_red[i]); s_red[0] = m; }
    __syncthreads();
    const float m = s_red[0];
    __syncthreads();

    float lsum = 0.0f;
    for (int e = t; e < len; e += 256) { float ev = __expf(s_coef[e] - m); s_coef[e] = ev; lsum += ev; }
    for (int off = 16; off; off >>= 1) lsum += __shfl_down(lsum, off, 32);
    if (lane == 0) s_red[wv] = lsum;
    __syncthreads();
    if (t == 0) { float s = 0.0f; for (int i = 0; i < 8; ++i) s += s_red[i]; s_red[0] = s; }
    __syncthreads();
    const float inv = 1.0f / s_red[0];
    __syncthreads();

    // Feature accumulation: 256 threads cover 512 features (2 each); per edge,
    // each wave reads a contiguous 128B slab of author_h (coalesced).
    const int f0 = t, f1 = t + 256;
    float acc0 = 0.0f, acc1 = 0.0f;
    for (int e = 0; e < len; ++e) {
        const float w = s_coef[e];                         // LDS broadcast
        const float* row = author_h + (size_t)s_src[e] * EMBED;
        acc0 += w * row[f0];
        acc1 += w * row[f1];
    }
    orow[f0] = acc0 * inv;
    orow[f1] = acc1 * inv;
}

// ---------------------------------------------------------------------------
// Kernel 2: paper_embs = accum_table[final_boxes] -> mix_in[:, 512:1024]
// ---------------------------------------------------------------------------
__global__ __launch_bounds__(256) void paper_fill(
    const float* __restrict__ accum_table, const int* __restrict__ boxes,
    float* __restrict__ mix_in) {
    const int idx = blockIdx.x * 256 + threadIdx.x;
    if (idx >= B_PAPERS * EMBED) return;
    const int p = idx >> 9;
    const int c = idx & 511;
    mix_in[(size_t)p * (2 * EMBED) + EMBED + c] = accum_table[(size_t)boxes[p] * EMBED + c];
}

// ---------------------------------------------------------------------------
// Kernel 3: C = lrelu(A[MxK] @ W[KxN] + bias[N]) via fp32 WMMA 16x16x4.
// 128x128 tile / workgroup (8 waves); each wave: 2x4 grid of 16x16 tiles.
// Tiles are streamed with CDNA5 async global->LDS copies (ASYNCcnt) into a
// double buffer: chunk k+1 is in flight while WMMAs consume chunk k.
// As stride 44 floats (176B: 16B-aligned rows for async B128 stores and
// conflict-free column reads); Bs stride 148 floats (592B, same properties).
// ---------------------------------------------------------------------------
__global__ __launch_bounds__(256) void gemm_bias_lrelu_wmma(
    const float* __restrict__ A, const float* __restrict__ W,
    const float* __restrict__ bias, float* __restrict__ C,
    int M, int N, int K) {
    __shared__ float As[2][128][44];
    __shared__ float Bs[2][32][148];

    const int t = threadIdx.x;
    const int lane = t & 31;
    const int wv = t >> 5;
    const int h = lane >> 4;     // half-wave select (K pair / M+8 select)
    const int lr = lane & 15;
    const int wm = wv >> 1;      // 0..3  -> 32-row band
    const int wn = wv & 1;       // 0..1  -> 64-col band
    const int m0 = blockIdx.y * 128;
    const int n0 = blockIdx.x * 128;

    // Issue one 32-deep K-chunk of A (128x32) and B (32x128) as 8 async B128
    // copies per thread-iteration (8 async instructions per wave per chunk).
    auto issue_chunk = [&](int buf, int kk) {
#pragma unroll
        for (int i = 0; i < 4; ++i) {             // A tile: 1024 float4 slots
            const int li = t + i * 256;
            const int row = li >> 3;
            const int c4 = (li & 7) << 2;
            async_copy_b128(lds_off32(&As[buf][row][c4]),
                            A + (size_t)(m0 + row) * K + kk + c4);
        }
#pragma unroll
        for (int i = 0; i < 4; ++i) {             // B tile: 1024 float4 slots
            const int li = t + i * 256;
            const int row = li >> 5;
            const int c4 = (li & 31) << 2;
            async_copy_b128(lds_off32(&Bs[buf][row][c4]),
                            W + (size_t)(kk + row) * N + n0 + c4);
        }
    };

    v8f acc[2][4];
    const v8f vzero = {0.f, 0.f, 0.f, 0.f, 0.f, 0.f, 0.f, 0.f};
#pragma unroll
    for (int i = 0; i < 2; ++i)
#pragma unroll
        for (int j = 0; j < 4; ++j) acc[i][j] = vzero;

    issue_chunk(0, 0);                            // prologue: chunk 0 -> buf 0

    for (int kk = 0; kk < K; kk += 32) {
        const int buf = (kk >> 5) & 1;
        if (kk + 32 < K) {
            issue_chunk(buf ^ 1, kk + 32);        // prefetch next chunk
            // Async loads retire in order: <=8 outstanding means the current
            // chunk's 8 copies have landed in LDS.
            asm volatile("s_wait_asynccnt 8" ::: "memory");
        } else {
            asm volatile("s_wait_asynccnt 0" ::: "memory");
        }
        __syncthreads();

#pragma unroll
        for (int k4 = 0; k4 < 8; ++k4) {
            const int kb = k4 * 4 + 2 * h;        // this lane's K pair base
            v2f a[2], b[4];
#pragma unroll
            for (int mt = 0; mt < 2; ++mt) {      // A 16x4 frag: lane=row, v0/v1=K pair
                const int r = 32 * wm + 16 * mt + lr;
                a[mt].x = As[buf][r][kb];
                a[mt].y = As[buf][r][kb + 1];
            }
#pragma unroll
            for (int nt = 0; nt < 4; ++nt) {      // B 4x16 frag: lane=col, v0/v1=K pair
                const int cidx = 64 * wn + 16 * nt + lr;
                b[nt].x = Bs[buf][kb][cidx];
                b[nt].y = Bs[buf][kb + 1][cidx];
            }
#pragma unroll
            for (int mt = 0; mt < 2; ++mt)
#pragma unroll
                for (int nt = 0; nt < 4; ++nt)
                    acc[mt][nt] = __builtin_amdgcn_wmma_f32_16x16x4_f32(
                        false, a[mt], false, b[nt], (short)0, acc[mt][nt], false, false);
        }
        __syncthreads();   // everyone done reading buf before it is refilled
    }

    // Epilogue: bias + leaky relu + store. C/D layout: VGPR r -> M = 8*h + r.
#pragma unroll
    for (int mt = 0; mt < 2; ++mt) {
#pragma unroll
        for (int nt = 0; nt < 4; ++nt) {
            const int col = n0 + 64 * wn + 16 * nt + lr;
            const float bv = bias[col];
            const int rbase = m0 + 32 * wm + 16 * mt + 8 * h;
#pragma unroll
            for (int r = 0; r < 8; ++r) {
                const float v = acc[mt][nt][r] + bv;
                C[(size_t)(rbase + r) * N + col] = lrelu_f(v);
            }
        }
    }
}

// ---------------------------------------------------------------------------
// Kernel 4: per-paper cosine-loss contribution + lrelu GEMV output.
// ---------------------------------------------------------------------------
__global__ __launch_bounds__(256) void finalize_k(
    const float* __restrict__ pop, const float* __restrict__ mix,
    const float* __restrict__ W_out, const float* __restrict__ b_out,
    float* __restrict__ out, float* __restrict__ loss) {
    __shared__ float s_red[4][8];
    const int p = blockIdx.x, t = threadIdx.x, lane = t & 31, wv = t >> 5;
    const float* pr = pop + (size_t)p * HIDDEN;
    const float* mr = mix + (size_t)p * HIDDEN;

    float d_pm = 0.f, n_a = 0.f, n_m = 0.f, d_ow = 0.f;
#pragma unroll
    for (int i = t; i < HIDDEN; i += 256) {
        const float a = pr[i], m = mr[i];
        d_pm += a * m;
        n_a += a * a;
        n_m += m * m;
        d_ow += a * W_out[i];
    }
    for (int off = 16; off; off >>= 1) {
        d_pm += __shfl_down(d_pm, off, 32);
        n_a  += __shfl_down(n_a, off, 32);
        n_m  += __shfl_down(n_m, off, 32);
        d_ow += __shfl_down(d_ow, off, 32);
    }
    if (lane == 0) { s_red[0][wv] = d_pm; s_red[1][wv] = n_a; s_red[2][wv] = n_m; s_red[3][wv] = d_ow; }
    __syncthreads();
    if (t == 0) {
        float sp = 0.f, sa = 0.f, sm = 0.f, so = 0.f;
        for (int i = 0; i < 8; ++i) { sp += s_red[0][i]; sa += s_red[1][i]; sm += s_red[2][i]; so += s_red[3][i]; }
        const float o = so + b_out[0];
        out[p] = lrelu_f(o);
        const float cs = sp / (fmaxf(sqrtf(sa), 1e-8f) * fmaxf(sqrtf(sm), 1e-8f));
        atomicAdd(loss, (1.0f - cs) * (1.0f / (float)B_PAPERS));
    }
}

__global__ void zero_loss(float* __restrict__ loss) {
    if (threadIdx.x == 0) *loss = 0.0f;
}

// ---------------------------------------------------------------------------
extern "C" void kernel_launch(void* const* d_in, const int* in_sizes, int n_in,
                              void* d_out, int out_size, void* d_ws, size_t ws_size,
                              hipStream_t stream) {
    const float* snapshot    = (const float*)d_in[0];   // [B, H]
    const float* author_h    = (const float*)d_in[1];   // [N_AUTH, EMBED]
    const float* accum_cit   = (const float*)d_in[2];   // [N_AUTH]
    const float* accum_table = (const float*)d_in[3];   // [BOX+1, EMBED]
    const float* W_enc       = (const float*)d_in[4];   // [H, H]
    const float* b_enc       = (const float*)d_in[5];   // [H]
    const float* W_mix       = (const float*)d_in[6];   // [2*EMBED, H]
    const float* b_mix       = (const float*)d_in[7];   // [H]
    const float* W_out       = (const float*)d_in[8];   // [H, 1]
    const float* b_out       = (const float*)d_in[9];   // [1]
    const int*   edge_src    = (const int*)d_in[10];    // [E]
    const int*   edge_dst    = (const int*)d_in[11];    // [E], sorted
    const int*   boxes       = (const int*)d_in[12];    // [B]

    // Output layout: out[B] | pop_embs[B*H] | pop_loss[1]
    float* out_pred = (float*)d_out;
    float* pop_embs = out_pred + B_PAPERS;
    float* loss_ptr = pop_embs + (size_t)B_PAPERS * HIDDEN;

    // Workspace: mix_in[B, 2*EMBED] | pop_mix[B, H]
    float* mix_in  = (float*)d_ws;
    float* pop_mix = mix_in + (size_t)B_PAPERS * (2 * EMBED);

    zero_loss<<<1, 32, 0, stream>>>(loss_ptr);
    edge_softmax_agg<<<B_PAPERS, 256, 0, stream>>>(author_h, accum_cit, edge_src, edge_dst, mix_in);
    paper_fill<<<(B_PAPERS * EMBED + 255) / 256, 256, 0, stream>>>(accum_table, boxes, mix_in);

    dim3 gemm_grid(HIDDEN / 128, B_PAPERS / 128);
    gemm_bias_lrelu_wmma<<<gemm_grid, 256, 0, stream>>>(snapshot, W_enc, b_enc, pop_embs,
                                                        B_PAPERS, HIDDEN, HIDDEN);
    gemm_bias_lrelu_wmma<<<gemm_grid, 256, 0, stream>>>(mix_in, W_mix, b_mix, pop_mix,
                                                        B_PAPERS, HIDDEN, 2 * EMBED);

    finalize_k<<<B_PAPERS, 256, 0, stream>>>(pop_embs, pop_mix, W_out, b_out, out_pred, loss_ptr);
}